// Net_67456756351379
// MI455X (gfx1250) — compile-verified
//
#include <hip/hip_runtime.h>
#include <hip/hip_bf16.h>
#include <cmath>

// ---------------------------------------------------------------------------
// Shapes (match reference): B=1024, J=K=16, BF=256, T=2401, box kernel 351.
// Output: d_out[0:16384]   = xs_decoded [1024,16]
//         d_out[16384:...] = out_shift  [1024,16]  (= xs_encoded)
// ---------------------------------------------------------------------------
#define BB    1024
#define TLEN  2401
#define KHALF 175          // (351-1)/2

typedef float v2f __attribute__((ext_vector_type(2)));
typedef float v8f __attribute__((ext_vector_type(8)));

// ---------------------------------------------------------------------------
// GEMM: Y[M,N] = act(X[M,K] @ W[N,K]^T + bias[N])  using V_WMMA_F32_16X16X4_F32
// One wave computes one 16x16 tile. 8 waves (256 threads) per block.
//
// A (16x4 f32) lane layout (ISA 7.12.2): lanes 0-15 = rows, VGPR0 holds K=0
// (lanes 0-15) / K=2 (lanes 16-31), VGPR1 holds K=1 / K=3.  B (4x16) mirrors
// this with lanes indexing N.  C/D: VGPR r, lane l -> M = r + 8*(l>=16),
// N = l & 15.
// ---------------------------------------------------------------------------
__global__ void gemm_wmma_f32(const float* __restrict__ X,
                              const float* __restrict__ W,
                              const float* __restrict__ bias,
                              float* __restrict__ Y,
                              int M, int N, int K, int relu, int tiles)
{
    const int wave = threadIdx.x >> 5;
    const int lane = threadIdx.x & 31;
    const int tile = blockIdx.x * 8 + wave;
    if (tile >= tiles) return;                 // wave-uniform; EXEC stays all-1s

    const int tilesN = N >> 4;
    const int tm = tile / tilesN;
    const int tn = tile % tilesN;

    const int r16 = lane & 15;                 // row of A / col of B
    const int khi = lane >> 4;                 // 0: K pair {0,1}; 1: K pair {2,3}

    const float* __restrict__ Arow = X + (size_t)(tm * 16 + r16) * K;
    const float* __restrict__ Brow = W + (size_t)(tn * 16 + r16) * K;

    v8f acc = {};
    for (int kk = 0; kk < K; kk += 4) {
        v2f a = *(const v2f*)(Arow + kk + 2 * khi);   // {K=2*khi, K=2*khi+1}
        v2f b = *(const v2f*)(Brow + kk + 2 * khi);
        acc = __builtin_amdgcn_wmma_f32_16x16x4_f32(
                  /*neg_a=*/false, a, /*neg_b=*/false, b,
                  /*c_mod=*/(short)0, acc, /*reuse_a=*/false, /*reuse_b=*/false);
    }

    const float bv = bias[tn * 16 + r16];
    #pragma unroll
    for (int r = 0; r < 8; ++r) {
        float v = acc[r] + bv;
        if (relu) v = fmaxf(v, 0.0f);
        Y[(size_t)(tm * 16 + khi * 8 + r) * N + tn * 16 + r16] = v;
    }
}

// ---------------------------------------------------------------------------
// find_offset: one wave32 per (b,k) sequence; 4 waves per 128-thread block.
// signal[i] = pulse(i) + noise[i]   (noise_scale == 1.0f exactly)
// heat[i] ∝ P[min(i+175,T-1)] - P[i-176]   (prefix-sum sliding window)
// argmax with first-occurrence tie-breaking; emit t[peak].
// ---------------------------------------------------------------------------
__global__ void find_offset_kernel(const float* __restrict__ noise,
                                   float* __restrict__ offv,
                                   float amp)
{
    __shared__ float P[4][TLEN];               // 4 * 2401 * 4B = 38.4 KB LDS
    const int wave = threadIdx.x >> 5;
    const int lane = threadIdx.x & 31;
    const int seq  = blockIdx.x * 4 + wave;    // 16384 sequences total

    const float* __restrict__ sig = noise + (size_t)seq * TLEN;
    float* Pw = P[wave];

    // inclusive prefix sum via wave-scan chunks of 32
    float carry = 0.0f;
    for (int base = 0; base < TLEN; base += 32) {
        const int i = base + lane;
        float v = 0.0f;
        if (i < TLEN) {
            const float t = -6.0f + 0.005f * (float)i;
            const float p = (fabsf(t) < 0.5f) ? amp : 0.0f;
            v = p + sig[i];
        }
        #pragma unroll
        for (int d = 1; d < 32; d <<= 1) {
            const float up = __shfl_up(v, d, 32);
            if (lane >= d) v += up;
        }
        if (i < TLEN) Pw[i] = carry + v;
        carry += __shfl(v, 31, 32);
    }
    __syncthreads();

    // windowed-sum argmax (first max wins, matching jnp.argmax)
    float best  = -3.402823466e38f;
    int   bestI = 0;
    for (int base = 0; base < TLEN; base += 32) {
        const int i = base + lane;
        if (i < TLEN) {
            int hi = i + KHALF; if (hi > TLEN - 1) hi = TLEN - 1;
            const int lo = i - KHALF - 1;
            const float w = Pw[hi] - (lo >= 0 ? Pw[lo] : 0.0f);
            if (w > best) { best = w; bestI = i; }   // strict >: keeps earliest
        }
    }
    #pragma unroll
    for (int d = 16; d >= 1; d >>= 1) {
        const float ov = __shfl_down(best, d, 32);
        const int   oi = __shfl_down(bestI, d, 32);
        if (ov > best || (ov == best && oi < bestI)) { best = ov; bestI = oi; }
    }
    if (lane == 0) offv[seq] = -6.0f + 0.005f * (float)bestI;
}

// ---------------------------------------------------------------------------
// shift + boundary wrap; also emits out_shift (= xs_encoded)
// ---------------------------------------------------------------------------
__global__ void shift_wrap_kernel(const float* __restrict__ xs_enc,
                                  const float* __restrict__ offv,
                                  float* __restrict__ xe,
                                  float* __restrict__ out_shift)
{
    const int i = blockIdx.x * blockDim.x + threadIdx.x;
    if (i >= BB * 16) return;
    const float xenc = xs_enc[i];
    out_shift[i] = xenc;
    float v = fminf(fmaxf(xenc, -5.0f), 5.0f) + offv[i];
    const float t0 = -6.0f;
    const float tl = -6.0f + 0.005f * 2400.0f;
    if (v > tl) v += (t0 - tl);
    if (v < t0) v += (tl - t0);
    xe[i] = v;
}

// ---------------------------------------------------------------------------
extern "C" void kernel_launch(void* const* d_in, const int* in_sizes, int n_in,
                              void* d_out, int out_size, void* d_ws, size_t ws_size,
                              hipStream_t stream)
{
    (void)in_sizes; (void)n_in; (void)out_size; (void)ws_size;

    const float* sample = (const float*)d_in[0];
    const float* noise  = (const float*)d_in[1];
    const float* W1  = (const float*)d_in[2];   const float* b1  = (const float*)d_in[3];
    const float* W2  = (const float*)d_in[4];   const float* b2  = (const float*)d_in[5];
    const float* W3  = (const float*)d_in[6];   const float* b3  = (const float*)d_in[7];
    const float* W1d = (const float*)d_in[8];   const float* b1d = (const float*)d_in[9];
    const float* W2d = (const float*)d_in[10];  const float* b2d = (const float*)d_in[11];
    const float* W3d = (const float*)d_in[12];  const float* b3d = (const float*)d_in[13];
    float* out = (float*)d_out;

    // workspace layout (floats): 3.2 MB total
    float* H1  = (float*)d_ws;            // [1024,256]  (shared enc/dec)
    float* H2  = H1 + BB * 256;           // [1024,512]
    float* xs  = H2 + BB * 512;           // [1024,16] xs_encoded
    float* off = xs + BB * 16;            // [1024,16] t[peak]
    float* xe  = off + BB * 16;           // [1024,16] xs_transmitted

    // pulse amplitude, bit-identical fp32 replay of the reference arithmetic
    int cnt = 0;
    for (int i = 0; i < TLEN; ++i) {
        float t = -6.0f + 0.005f * (float)i;
        if (fabsf(t) < 0.5f) cnt++;
    }
    const float amp = 4.0f / ((float)cnt * 0.005f);   // sqrt(E0*J)=4, noise_scale=1

    auto gemm = [&](const float* X, const float* W, const float* bias, float* Y,
                    int M, int N, int K, int relu) {
        const int tiles = (M / 16) * (N / 16);
        gemm_wmma_f32<<<(tiles + 7) / 8, 256, 0, stream>>>(X, W, bias, Y,
                                                           M, N, K, relu, tiles);
    };

    // HBM-bound stage first (157 MB of noise, read exactly once)
    find_offset_kernel<<<(BB * 16) / 4, 128, 0, stream>>>(noise, off, amp);

    // encoder MLP (fp32 WMMA)
    gemm(sample, W1, b1, H1, BB, 256,  16, 1);
    gemm(H1,     W2, b2, H2, BB, 512, 256, 1);
    gemm(H2,     W3, b3, xs, BB,  16, 512, 0);

    // shift + wrap; also write out_shift into second half of d_out
    shift_wrap_kernel<<<(BB * 16 + 255) / 256, 256, 0, stream>>>(
        xs, off, xe, out + BB * 16);

    // decoder MLP (fp32 WMMA), final layer straight into d_out
    gemm(xe, W1d, b1d, H1, BB, 256,  16, 1);
    gemm(H1, W2d, b2d, H2, BB, 512, 256, 1);
    gemm(H2, W3d, b3d, out, BB,  16, 512, 0);
}